// SepConv4d_53704271069360
// MI455X (gfx1250) — compile-verified
//
#include <hip/hip_runtime.h>

#define BATCH 8
#define CI1   16     // cin of stage A conv (conv2)
#define COUT  32
#define S     24
#define P     576    // S*S
#define PADW  26
#define PADP  676    // 26*26
#define K2    144    // CI1*9   (stage A reduction)
#define K1    288    // COUT*9  (stage C reduction)

typedef float v2f __attribute__((ext_vector_type(2)));
typedef float v8f __attribute__((ext_vector_type(8)));

// ---------------------------------------------------------------------------
// zero group-norm stats accumulators (64 floats)
// ---------------------------------------------------------------------------
__global__ void zero_stats_kernel(float* s) {
    int t = threadIdx.x;
    if (t < 64) s[t] = 0.f;
}

// ---------------------------------------------------------------------------
// Stage A: conv over (hb,wb) plane for one (b,ha,wa) image.
//   GEMM: M=COUT(32), N=P(576), K=144, fp32 WMMA 16x16x4.
//   Writes Y laid out as [b][hawa][cout][hbwb] (stage-C friendly) and
//   accumulates per-(b,group) sum / sum-of-squares for GroupNorm.
// ---------------------------------------------------------------------------
__global__ __launch_bounds__(256) void conv_hbwb_kernel(
    const float* __restrict__ x,   // (B, CI1, S,S, S,S)
    const float* __restrict__ w2,  // (COUT, CI1, 3, 3)
    const float* __restrict__ b2,  // (COUT)
    float* __restrict__ Y,         // (B, P, COUT, P)
    float* __restrict__ gsum,      // (B*4)
    float* __restrict__ gsq)       // (B*4)
{
    __shared__ float xs[CI1 * PADP];   // padded input plane  (43.3 KB)
    __shared__ float ws[COUT * K2];    // weights             (18.4 KB)
    __shared__ int   koff[K2];
    __shared__ float bias[COUT];
    __shared__ float lsum[4], lsq[4];

    const int blk  = blockIdx.x;          // b*P + hawa
    const int b    = blk / P;
    const int hawa = blk % P;
    const int t    = threadIdx.x;

    for (int i = t; i < COUT * K2; i += 256) ws[i] = w2[i];
    if (t < K2) {
        int cin = t / 9, r = t % 9;
        koff[t] = cin * PADP + (r / 3) * PADW + (r % 3);
    }
    if (t < COUT) bias[t] = b2[t];
    if (t < 4) { lsum[t] = 0.f; lsq[t] = 0.f; }

    // padded load of x[b][cin][hawa][*]: consecutive threads -> consecutive hbwb
    for (int i = t; i < CI1 * PADP; i += 256) {
        int cin = i / PADP, r = i % PADP;
        int hh = r / PADW, ww = r % PADW;
        int hb = hh - 1, wb = ww - 1;
        float v = 0.f;
        if ((unsigned)hb < (unsigned)S && (unsigned)wb < (unsigned)S)
            v = x[(((size_t)b * CI1 + cin) * P + hawa) * P + hb * S + wb];
        xs[i] = v;
    }
    __syncthreads();

    const int lane = t & 31, wave = t >> 5;
    const int half = lane >> 4, lq = lane & 15;

    // 2 M-tiles x 36 N-tiles = 72 tiles; 8 waves -> 9 tiles each (uniform)
    for (int tile = wave; tile < 72; tile += 8) {
        const int mtile = tile & 1;
        const int n0    = (tile >> 1) << 4;
        const int n     = n0 + lq;
        const int nbase = (n / S) * PADW + (n % S);
        const int m     = (mtile << 4) + lq;

        v8f acc = {0.f, 0.f, 0.f, 0.f, 0.f, 0.f, 0.f, 0.f};
        #pragma unroll 4
        for (int k0 = 0; k0 < K2; k0 += 4) {
            const int ka = k0 + (half << 1);
            v2f av = { ws[m * K2 + ka], ws[m * K2 + ka + 1] };
            v2f bv = { xs[koff[ka] + nbase], xs[koff[ka + 1] + nbase] };
            acc = __builtin_amdgcn_wmma_f32_16x16x4_f32(
                false, av, false, bv, (short)0, acc, false, false);
        }

        // D layout: lane = column n; rows = vgpr + 8*half. All 8 values of a
        // lane share one GN group: g = mtile*2 + half.
        float s = 0.f, q = 0.f;
        float* yp = Y + ((size_t)b * P + hawa) * COUT * P;
        #pragma unroll
        for (int i = 0; i < 8; i++) {
            int mm = (mtile << 4) + (half << 3) + i;
            float v = acc[i] + bias[mm];
            yp[(size_t)mm * P + n0 + lq] = v;
            s += v; q += v * v;
        }
        const int g = (mtile << 1) + half;
        atomicAdd(&lsum[g], s);
        atomicAdd(&lsq[g], q);
    }
    __syncthreads();
    if (t < 4) {
        atomicAdd(&gsum[b * 4 + t], lsum[t]);
        atomicAdd(&gsq[b * 4 + t], lsq[t]);
    }
}

// ---------------------------------------------------------------------------
// GroupNorm stats -> per-(b,c) affine scale/shift (folds gamma/beta/mean/var)
// ---------------------------------------------------------------------------
__global__ void stats_kernel(const float* __restrict__ gsum,
                             const float* __restrict__ gsq,
                             const float* __restrict__ gamma,
                             const float* __restrict__ beta,
                             float* __restrict__ scale,
                             float* __restrict__ shift)
{
    int t = threadIdx.x;           // 256 = B*COUT
    int b = t >> 5, c = t & 31, g = c >> 3;
    const float cnt  = (float)(8 * P * P);   // 8 chans * 576 * 576
    float mean = gsum[b * 4 + g] / cnt;
    float var  = gsq [b * 4 + g] / cnt - mean * mean;
    float inv  = rsqrtf(var + 1e-5f);
    float sc   = gamma[c] * inv;
    scale[t] = sc;
    shift[t] = beta[c] - mean * sc;
}

// ---------------------------------------------------------------------------
// Stage C: conv over (ha,wa) plane. Block = (b, 8-pixel hbwb chunk, 2-row ha
// band). Normalize+ReLU fused into the LDS fill. WMMA columns = 2 wa x 8 pix.
//   GEMM per block: M=32, N=2(ha)x24(wa)x8(pix), K=288.
// Dynamic LDS: ys 4*26*32*8 + w 32*288 + tables  = ~145 KB (<320 KB WGP LDS).
// ---------------------------------------------------------------------------
__global__ __launch_bounds__(256) void conv_hawa_kernel(
    const float* __restrict__ Y,     // (B, P, COUT, P)
    const float* __restrict__ w1,    // (COUT, COUT, 3, 3)
    const float* __restrict__ b1,    // (COUT)
    const float* __restrict__ scale, // (B*COUT)
    const float* __restrict__ shift, // (B*COUT)
    float* __restrict__ out)         // (B, COUT, ha,wa, hb,wb)
{
    extern __shared__ float smem[];
    float* ys   = smem;                    // [4][26][32][8]
    float* ws   = ys + 4 * PADW * COUT * 8;// [32][288]
    float* bias = ws + COUT * K1;          // [32]
    float* ss   = bias + COUT;             // [32]
    float* sh   = ss + COUT;               // [32]
    int*   koff = (int*)(sh + COUT);       // [288]

    const int t    = threadIdx.x;
    const int blk  = blockIdx.x;           // b*(72*12) + chunk*12 + band
    const int b    = blk / (72 * 12);
    const int r    = blk % (72 * 12);
    const int pix0 = (r / 12) * 8;
    const int h0   = (r % 12) * 2;

    for (int i = t; i < COUT * K1; i += 256) ws[i] = w1[i];
    if (t < K1) {
        int cin = t / 9, rr = t % 9;
        koff[t] = (((rr / 3) * PADW + (rr % 3)) * COUT + cin) * 8;
    }
    if (t < COUT) {
        bias[t] = b1[t];
        ss[t]   = scale[b * COUT + t];
        sh[t]   = shift[b * COUT + t];
    }
    __syncthreads();

    // padded, normalized, ReLU'd input band; consecutive threads -> contiguous
    for (int i = t; i < 4 * PADW * COUT * 8; i += 256) {
        int p   = i & 7;
        int t2  = i >> 3;
        int cin = t2 & 31;
        int t3  = t2 >> 5;
        int ww  = t3 % PADW;
        int hh  = t3 / PADW;
        int ha  = h0 + hh - 1, wa = ww - 1;
        float v = 0.f;
        if ((unsigned)ha < (unsigned)S && (unsigned)wa < (unsigned)S) {
            float y = Y[(((size_t)b * P + ha * S + wa) * COUT + cin) * P + pix0 + p];
            v = fmaxf(y * ss[cin] + sh[cin], 0.f);
        }
        ys[i] = v;
    }
    __syncthreads();

    const int lane = t & 31, wave = t >> 5;
    const int half = lane >> 4, lq = lane & 15;
    const int p    = lq & 7,  wd  = lq >> 3;

    // 2 M-tiles x 2 ha rows x 12 wa-pairs = 48 tiles; 8 waves -> 6 each
    for (int tile = wave; tile < 48; tile += 8) {
        const int mtile = tile & 1;
        const int rem   = tile >> 1;
        const int har   = rem & 1;
        const int wa0   = (rem >> 1) << 1;
        const int m     = (mtile << 4) + lq;
        const int cbase = ((har * PADW + wa0 + wd) * COUT) * 8 + p;

        v8f acc = {0.f, 0.f, 0.f, 0.f, 0.f, 0.f, 0.f, 0.f};
        #pragma unroll 4
        for (int k0 = 0; k0 < K1; k0 += 4) {
            const int ka = k0 + (half << 1);
            v2f av = { ws[m * K1 + ka], ws[m * K1 + ka + 1] };
            v2f bv = { ys[koff[ka] + cbase], ys[koff[ka + 1] + cbase] };
            acc = __builtin_amdgcn_wmma_f32_16x16x4_f32(
                false, av, false, bv, (short)0, acc, false, false);
        }

        const int ha = h0 + har;
        #pragma unroll
        for (int i = 0; i < 8; i++) {
            int mm = (mtile << 4) + (half << 3) + i;
            float v = acc[i] + bias[mm];
            out[(((size_t)b * COUT + mm) * P + ha * S + wa0 + wd) * P + pix0 + p] = v;
        }
    }
}

// ---------------------------------------------------------------------------
extern "C" void kernel_launch(void* const* d_in, const int* in_sizes, int n_in,
                              void* d_out, int out_size, void* d_ws, size_t ws_size,
                              hipStream_t stream) {
    (void)in_sizes; (void)n_in; (void)out_size; (void)ws_size;
    const float* x     = (const float*)d_in[0];
    const float* w1    = (const float*)d_in[1];
    const float* b1    = (const float*)d_in[2];
    const float* w2    = (const float*)d_in[3];
    const float* b2    = (const float*)d_in[4];
    const float* gamma = (const float*)d_in[5];
    const float* beta  = (const float*)d_in[6];
    float* out = (float*)d_out;

    float* Y     = (float*)d_ws;                       // 340 MB intermediate
    const size_t YN = (size_t)BATCH * P * COUT * P;
    float* gsum  = Y + YN;                             // 32
    float* gsq   = gsum + 32;                          // 32
    float* scale = gsq + 32;                           // 256
    float* shift = scale + 256;                        // 256

    zero_stats_kernel<<<1, 64, 0, stream>>>(gsum);
    conv_hbwb_kernel<<<BATCH * P, 256, 0, stream>>>(x, w2, b2, Y, gsum, gsq);
    stats_kernel<<<1, 256, 0, stream>>>(gsum, gsq, gamma, beta, scale, shift);

    const size_t smem = (size_t)(4 * PADW * COUT * 8 + COUT * K1 + 3 * COUT) * sizeof(float)
                      + (size_t)K1 * sizeof(int);
    conv_hawa_kernel<<<BATCH * 72 * 12, 256, smem, stream>>>(Y, w1, b1, scale, shift, out);
}